// GNNInfer_82008105549935
// MI455X (gfx1250) — compile-verified
//
#include <hip/hip_runtime.h>
#include <hip/hip_bf16.h>

typedef __attribute__((ext_vector_type(2))) float v2f;
typedef __attribute__((ext_vector_type(8))) float v8f;

#define HDIM 128
#define ZDIM 64
#define INDIM 6
#define BIN 198
#define BINP 200   // K padded to multiple of 4 (cols 198,199 zero)

// ---------------------------------------------------------------------------
// prep: build in6 = [vertices | vector_potential], zero deg and msg6
// ---------------------------------------------------------------------------
__global__ __launch_bounds__(256) void prep_kernel(
    const float* __restrict__ vert, const float* __restrict__ vp,
    float* __restrict__ in6, float* __restrict__ deg,
    float* __restrict__ msg6, int n)
{
    int i = blockIdx.x * blockDim.x + threadIdx.x;
    if (i >= n) return;
    in6[i * 6 + 0] = vert[i * 3 + 0];
    in6[i * 6 + 1] = vert[i * 3 + 1];
    in6[i * 6 + 2] = vert[i * 3 + 2];
    in6[i * 6 + 3] = vp[i * 3 + 0];
    in6[i * 6 + 4] = vp[i * 3 + 1];
    in6[i * 6 + 5] = vp[i * 3 + 2];
    deg[i] = 0.0f;
#pragma unroll
    for (int k = 0; k < 6; ++k) msg6[i * 6 + k] = 0.0f;
}

// ---------------------------------------------------------------------------
// per-edge: degree count + 6-wide message scatter (layer 0 aggregation)
// ---------------------------------------------------------------------------
__global__ __launch_bounds__(256) void edge_deg_msg6_kernel(
    const int2* __restrict__ edges, const float* __restrict__ in6,
    float* __restrict__ deg, float* __restrict__ msg6, int e)
{
    int t = blockIdx.x * blockDim.x + threadIdx.x;
    if (t >= e) return;
    int2 ed = edges[t];
    int src = ed.x, dst = ed.y;
    atomicAdd(&deg[dst], 1.0f);
#pragma unroll
    for (int k = 0; k < 6; ++k)
        atomicAdd(&msg6[dst * 6 + k], in6[src * 6 + k]);
}

__global__ __launch_bounds__(256) void invdeg_kernel(float* __restrict__ deg, int n)
{
    int i = blockIdx.x * blockDim.x + threadIdx.x;
    if (i >= n) return;
    deg[i] = 1.0f / fmaxf(deg[i], 1.0f);
}

// ---------------------------------------------------------------------------
// layer 0: h = relu(in6 @ W0s + (msg6*invdeg) @ W0n + b0)   (K=6, VALU)
// one block of 128 threads per node
// ---------------------------------------------------------------------------
__global__ __launch_bounds__(128) void layer0_kernel(
    const float* __restrict__ in6, const float* __restrict__ msg6,
    const float* __restrict__ invdeg,
    const float* __restrict__ W0s, const float* __restrict__ W0n,
    const float* __restrict__ b0, float* __restrict__ h, int n)
{
    int i = blockIdx.x;
    int c = threadIdx.x;
    if (i >= n) return;
    float id = invdeg[i];
    float acc = b0[c];
#pragma unroll
    for (int k = 0; k < 6; ++k) {
        float xs = in6[i * 6 + k];
        float xn = msg6[i * 6 + k] * id;
        acc += xs * W0s[k * HDIM + c] + xn * W0n[k * HDIM + c];
    }
    h[i * HDIM + c] = fmaxf(acc, 0.0f);
}

// ---------------------------------------------------------------------------
// pad a layer's Ws/Wn (198x128) into zero-padded 200x128 workspace copies
// (removes all bounds checks from the hot GEMM loop)
// ---------------------------------------------------------------------------
__global__ __launch_bounds__(256) void padW_kernel(
    const float* __restrict__ Ws, const float* __restrict__ Wn,
    float* __restrict__ Wsp, float* __restrict__ Wnp)
{
    int t = blockIdx.x * blockDim.x + threadIdx.x;
    if (t >= BINP * HDIM) return;
    int k = t >> 7;               // row
    float vs = 0.0f, vn = 0.0f;
    if (k < BIN) { vs = Ws[t]; vn = Wn[t]; }
    Wsp[t] = vs;
    Wnp[t] = vn;
}

// ---------------------------------------------------------------------------
// Fused dual GEMM:
//   ys[N,128] = Xcat @ Wsp ;  yn[N,128] = Xcat @ Wnp
// where Xcat = concat(h[N,128], z[N,64], in6[N,6]) (K padded to 200).
// f32 WMMA 16x16x4. One 16-row X tile per block staged in LDS (loaded ONCE
// for both matmuls); 8 waves = 8 column tiles; 2 accumulators per wave.
// ---------------------------------------------------------------------------
__global__ __launch_bounds__(256) void gemm198_dual_wmma_kernel(
    const float* __restrict__ h, const float* __restrict__ zi,
    const float* __restrict__ in6,
    const float* __restrict__ Wsp, const float* __restrict__ Wnp,
    float* __restrict__ ys, float* __restrict__ yn, int n)
{
    __shared__ float Xs[16 * BINP];

    int tid = threadIdx.x;
    int rowBase = blockIdx.x * 16;

    // cooperative load of the concatenated 16 x 200 X tile (zero padded)
    for (int idx = tid; idx < 16 * BINP; idx += 256) {
        int r = idx / BINP;
        int c = idx - r * BINP;
        int row = rowBase + r;
        float v = 0.0f;
        if (row < n) {
            if (c < HDIM)             v = h[row * HDIM + c];
            else if (c < HDIM + ZDIM) v = zi[row * ZDIM + (c - HDIM)];
            else if (c < BIN)         v = in6[row * INDIM + (c - HDIM - ZDIM)];
        }
        Xs[idx] = v;
    }
    __syncthreads();

    int wave = tid >> 5;            // 0..7 -> output column tile
    int lane = tid & 31;
    int col0 = wave * 16;
    int m     = lane & 15;          // A row within tile
    int khalf = (lane >> 4) * 2;    // 0 or 2 : K sub-offset per half-wave
    int nn    = lane & 15;          // B/C column within tile

    const float* ps = &Wsp[khalf * HDIM + col0 + nn];
    const float* pn = &Wnp[khalf * HDIM + col0 + nn];

    v8f accS = {};
    v8f accN = {};
    for (int k = 0; k < BINP; k += 4) {
        // A fragment: lane holds X[m][k+khalf], X[m][k+khalf+1] (8B aligned)
        v2f a = *(const v2f*)(&Xs[m * BINP + k + khalf]);
        // B fragments (unguarded: weights are pre-padded to 200 rows)
        v2f bs; bs.x = ps[k * HDIM]; bs.y = ps[(k + 1) * HDIM];
        v2f bn; bn.x = pn[k * HDIM]; bn.y = pn[(k + 1) * HDIM];
        accS = __builtin_amdgcn_wmma_f32_16x16x4_f32(
            false, a, false, bs, (short)0, accS, false, false);
        accN = __builtin_amdgcn_wmma_f32_16x16x4_f32(
            false, a, false, bn, (short)0, accN, false, false);
    }

    // C layout: VGPR r -> M = r (lanes 0-15) / M = r+8 (lanes 16-31), N = nn
    int mOut = (lane >> 4) * 8;
    int base = (rowBase + mOut) * HDIM + col0 + nn;
    if (rowBase + 16 <= n) {
        // full tile: unguarded straight-line stores
#pragma unroll
        for (int r = 0; r < 8; ++r) {
            ys[base + r * HDIM] = accS[r];
            yn[base + r * HDIM] = accN[r];
        }
    } else {
#pragma unroll
        for (int r = 0; r < 8; ++r) {
            int row = rowBase + mOut + r;
            if (row < n) {
                ys[row * HDIM + col0 + nn] = accS[r];
                yn[row * HDIM + col0 + nn] = accN[r];
            }
        }
    }
}

// ---------------------------------------------------------------------------
// scatter: ys[dst] += yn[src] * invdeg[dst]   (128 floats/edge, 32 thr/edge)
// ---------------------------------------------------------------------------
__global__ __launch_bounds__(256) void scatter128_kernel(
    const int2* __restrict__ edges, const float* __restrict__ yn,
    const float* __restrict__ invdeg, float* __restrict__ ys, int e)
{
    long long t = (long long)blockIdx.x * blockDim.x + threadIdx.x;
    int edge = (int)(t >> 5);
    if (edge >= e) return;
    int k4 = ((int)t & 31) * 4;
    int2 ed = edges[edge];
    int src = ed.x, dst = ed.y;
    float s = invdeg[dst];
    const float4 v = *(const float4*)(&yn[src * HDIM + k4]);
    float* p = &ys[dst * HDIM + k4];
    atomicAdd(p + 0, v.x * s);
    atomicAdd(p + 1, v.y * s);
    atomicAdd(p + 2, v.z * s);
    atomicAdd(p + 3, v.w * s);
}

// ---------------------------------------------------------------------------
// bias + relu (in place)
// ---------------------------------------------------------------------------
__global__ __launch_bounds__(256) void bias_relu_kernel(
    float* __restrict__ ys, const float* __restrict__ b, int n)
{
    int t = blockIdx.x * blockDim.x + threadIdx.x;
    if (t >= n * HDIM) return;
    int c = t & (HDIM - 1);
    float v = ys[t] + b[c];
    ys[t] = v > 0.0f ? v : 0.0f;
}

// ---------------------------------------------------------------------------
// final layer GEMM: out = h @ Wls + bl ; yn3 = h @ Wln  (one wave per node)
// ---------------------------------------------------------------------------
__global__ __launch_bounds__(256) void final_gemm_kernel(
    const float* __restrict__ h, const float* __restrict__ Wls,
    const float* __restrict__ Wln, const float* __restrict__ bl,
    float* __restrict__ out, float* __restrict__ yn3, int n)
{
    int t = blockIdx.x * blockDim.x + threadIdx.x;
    int i = t >> 5;
    int lane = t & 31;
    if (i >= n) return;
    float ss0 = 0.f, ss1 = 0.f, ss2 = 0.f;
    float sn0 = 0.f, sn1 = 0.f, sn2 = 0.f;
    for (int c = lane; c < HDIM; c += 32) {
        float x = h[i * HDIM + c];
        ss0 += x * Wls[c * 3 + 0];
        ss1 += x * Wls[c * 3 + 1];
        ss2 += x * Wls[c * 3 + 2];
        sn0 += x * Wln[c * 3 + 0];
        sn1 += x * Wln[c * 3 + 1];
        sn2 += x * Wln[c * 3 + 2];
    }
#pragma unroll
    for (int off = 16; off > 0; off >>= 1) {
        ss0 += __shfl_down(ss0, off, 32);
        ss1 += __shfl_down(ss1, off, 32);
        ss2 += __shfl_down(ss2, off, 32);
        sn0 += __shfl_down(sn0, off, 32);
        sn1 += __shfl_down(sn1, off, 32);
        sn2 += __shfl_down(sn2, off, 32);
    }
    if (lane == 0) {
        out[i * 3 + 0] = ss0 + bl[0];
        out[i * 3 + 1] = ss1 + bl[1];
        out[i * 3 + 2] = ss2 + bl[2];
        yn3[i * 3 + 0] = sn0;
        yn3[i * 3 + 1] = sn1;
        yn3[i * 3 + 2] = sn2;
    }
}

__global__ __launch_bounds__(256) void final_scatter_kernel(
    const int2* __restrict__ edges, const float* __restrict__ yn3,
    const float* __restrict__ invdeg, float* __restrict__ out, int e)
{
    int t = blockIdx.x * blockDim.x + threadIdx.x;
    if (t >= e) return;
    int2 ed = edges[t];
    int src = ed.x, dst = ed.y;
    float s = invdeg[dst];
#pragma unroll
    for (int k = 0; k < 3; ++k)
        atomicAdd(&out[dst * 3 + k], yn3[src * 3 + k] * s);
}

// ---------------------------------------------------------------------------
extern "C" void kernel_launch(void* const* d_in, const int* in_sizes, int n_in,
                              void* d_out, int out_size, void* d_ws, size_t ws_size,
                              hipStream_t stream)
{
    const float* vertices = (const float*)d_in[0];   // N x 3
    const int2*  edges    = (const int2*) d_in[1];   // E x 2 (int32 pairs)
    const float* vp       = (const float*)d_in[2];   // N x 3
    const float* z        = (const float*)d_in[3];   // 4 x N x 64
    const float* W0s      = (const float*)d_in[4];   // 6 x 128
    const float* W0n      = (const float*)d_in[5];   // 6 x 128
    const float* b0       = (const float*)d_in[6];   // 128
    const float* Wbs      = (const float*)d_in[7];   // 4 x 198 x 128
    const float* Wbn      = (const float*)d_in[8];   // 4 x 198 x 128
    const float* bb       = (const float*)d_in[9];   // 4 x 128
    const float* Wls      = (const float*)d_in[10];  // 128 x 3
    const float* Wln      = (const float*)d_in[11];  // 128 x 3
    const float* bl       = (const float*)d_in[12];  // 3
    float* out = (float*)d_out;

    const int N = in_sizes[0] / 3;
    const int E = in_sizes[1] / 2;

    // workspace carve-up (256B aligned slices)
    char* ws = (char*)d_ws;
    size_t off = 0;
    auto carve = [&](size_t bytes) -> float* {
        float* p = (float*)(ws + off);
        off = (off + bytes + 255) & ~(size_t)255;
        return p;
    };
    float* invdeg = carve((size_t)N * 4);
    float* in6    = carve((size_t)N * 6 * 4);
    float* msg6   = carve((size_t)N * 6 * 4);   // reused as yn3 in final layer
    float* Wsp    = carve((size_t)BINP * HDIM * 4);  // padded weights (per layer)
    float* Wnp    = carve((size_t)BINP * HDIM * 4);
    float* bufA   = carve((size_t)N * HDIM * 4);
    float* bufB   = carve((size_t)N * HDIM * 4);
    float* bufC   = carve((size_t)N * HDIM * 4);
    (void)ws_size; (void)n_in; (void)out_size;

    const int TB = 256;
    int gN   = (N + TB - 1) / TB;
    int gE   = (E + TB - 1) / TB;
    int gNH  = (int)(((size_t)N * HDIM + TB - 1) / TB);
    int gE32 = (int)(((size_t)E * 32 + TB - 1) / TB);
    int gN32 = (int)(((size_t)N * 32 + TB - 1) / TB);
    int gTile = (N + 15) / 16;
    int gPad  = (BINP * HDIM + TB - 1) / TB;

    // --- degree + layer-0 aggregation ---
    prep_kernel<<<gN, TB, 0, stream>>>(vertices, vp, in6, invdeg, msg6, N);
    edge_deg_msg6_kernel<<<gE, TB, 0, stream>>>(edges, in6, invdeg, msg6, E);
    invdeg_kernel<<<gN, TB, 0, stream>>>(invdeg, N);

    // --- layer 0 ---
    layer0_kernel<<<N, HDIM, 0, stream>>>(in6, msg6, invdeg, W0s, W0n, b0, bufA, N);

    // --- 4 hidden graph-conv layers (fused dual WMMA GEMM, then aggregate) ---
    float* hcur = bufA;
    float* ybuf = bufB;
    for (int i = 0; i < 4; ++i) {
        const float* zi  = z   + (size_t)i * N * ZDIM;
        const float* Wsi = Wbs + (size_t)i * BIN * HDIM;
        const float* Wni = Wbn + (size_t)i * BIN * HDIM;
        const float* bi  = bb  + (size_t)i * HDIM;

        padW_kernel<<<gPad, TB, 0, stream>>>(Wsi, Wni, Wsp, Wnp);
        gemm198_dual_wmma_kernel<<<gTile, TB, 0, stream>>>(
            hcur, zi, in6, Wsp, Wnp, ybuf, bufC, N);
        scatter128_kernel<<<gE32, TB, 0, stream>>>(edges, bufC, invdeg, ybuf, E);
        bias_relu_kernel<<<gNH, TB, 0, stream>>>(ybuf, bi, N);

        float* tmp = hcur; hcur = ybuf; ybuf = tmp;
    }

    // --- final layer (K=128 -> 3), aggregate after GEMM over only 3 dims ---
    float* yn3 = msg6;  // N x 3, reuse
    final_gemm_kernel<<<gN32, TB, 0, stream>>>(hcur, Wls, Wln, bl, out, yn3, N);
    final_scatter_kernel<<<gE, TB, 0, stream>>>(edges, yn3, invdeg, out, E);
}